// SimplifiedDRNLayer_62483184222676
// MI455X (gfx1250) — compile-verified
//
#include <hip/hip_runtime.h>

#define NUM_POPS 8
#define IN_DIM   2048
#define OUT_DIM  2048
#define BATCH    8192

// GEMM tiling
#define TM   64            // samples per block tile (4 WMMA row-tiles)
#define TNB  128           // output cols per block (8 waves x 16)
#define KC   128           // K chunk staged in LDS (double-buffered)
#define LDSS (KC + 4)      // padded row stride (floats) -> conflict-free ds_load_b64
#define NCHUNK (IN_DIM / KC)
#define BUFSZ (TM * LDSS)  // floats per LDS buffer
#define ISSUES 8           // async B128 issues per thread per chunk (KC/4/4)

#define AS1 __attribute__((address_space(1)))
#define AS3 __attribute__((address_space(3)))

typedef __attribute__((ext_vector_type(2))) float v2f;
typedef __attribute__((ext_vector_type(8))) float v8f;
typedef __attribute__((ext_vector_type(4))) int   v4i;

// ---------------------------------------------------------------------------
// Kernel 1: zero output + expert counters (d_out/d_ws are poisoned by harness)
// ---------------------------------------------------------------------------
__global__ void zero_kernel(float* __restrict__ out, int n, int* __restrict__ counts) {
    int tid = blockIdx.x * blockDim.x + threadIdx.x;
    if (tid < NUM_POPS) counts[tid] = 0;
    float4 z = make_float4(0.f, 0.f, 0.f, 0.f);
    float4* o4 = (float4*)out;
    int n4 = n >> 2;
    for (int i = tid; i < n4; i += gridDim.x * blockDim.x) o4[i] = z;
}

// ---------------------------------------------------------------------------
// Kernel 2: gating. One wave (32 lanes) per sample: 8 dot products of length
// 2048, top-2 of logits (softmax is monotonic -> skip it), append sample id
// to the two selected experts' lists.
// ---------------------------------------------------------------------------
__global__ void gate_kernel(const float* __restrict__ x,
                            const float* __restrict__ Ws,
                            const float* __restrict__ bs,
                            int* __restrict__ counts,
                            int* __restrict__ lists) {
    int wave = threadIdx.x >> 5;
    int lane = threadIdx.x & 31;
    int b = blockIdx.x * 8 + wave;    // 8 waves per block
    if (b >= BATCH) return;

    const float4* x4 = (const float4*)(x + (size_t)b * IN_DIM);
    const float4* w4 = (const float4*)Ws;

    float acc[NUM_POPS];
#pragma unroll
    for (int p = 0; p < NUM_POPS; ++p) acc[p] = 0.f;

    for (int i = lane; i < IN_DIM / 4; i += 32) {
        float4 xv = x4[i];
#pragma unroll
        for (int p = 0; p < NUM_POPS; ++p) {
            float4 wv = w4[p * (IN_DIM / 4) + i];
            acc[p] += xv.x * wv.x + xv.y * wv.y + xv.z * wv.z + xv.w * wv.w;
        }
    }
#pragma unroll
    for (int p = 0; p < NUM_POPS; ++p) {
#pragma unroll
        for (int off = 16; off > 0; off >>= 1)
            acc[p] += __shfl_xor(acc[p], off, 32);
    }

    if (lane == 0) {
        float lg[NUM_POPS];
#pragma unroll
        for (int p = 0; p < NUM_POPS; ++p) lg[p] = acc[p] + bs[p];
        // top-2, ties -> lowest index (matches jax.lax.top_k)
        int i1 = 0;
#pragma unroll
        for (int p = 1; p < NUM_POPS; ++p) if (lg[p] > lg[i1]) i1 = p;
        int i2 = (i1 == 0) ? 1 : 0;
#pragma unroll
        for (int p = 0; p < NUM_POPS; ++p)
            if (p != i1 && lg[p] > lg[i2]) i2 = p;

        int pos1 = atomicAdd(&counts[i1], 1);
        lists[i1 * BATCH + pos1] = b;
        int pos2 = atomicAdd(&counts[i2], 1);
        lists[i2 * BATCH + pos2] = b;
    }
}

// ---------------------------------------------------------------------------
// Kernel 3: grouped expert GEMM using V_WMMA_F32_16X16X4_F32 (full f32).
// A tile (gathered x rows) staged via double-buffered async global->LDS
// copies (ASYNCcnt path); B fragments streamed from W as float2.
// out += 0.5*(W[p]x + b[p]) via global f32 atomics (2 experts per sample).
// ---------------------------------------------------------------------------
__launch_bounds__(256)
__global__ void moe_gemm_kernel(const float* __restrict__ x,
                                const float* __restrict__ W,
                                const float* __restrict__ bias,
                                const int*   __restrict__ counts,
                                const int*   __restrict__ lists,
                                float* __restrict__ out) {
    int p = blockIdx.z;
    int cnt = counts[p];
    int tile0 = blockIdx.x * TM;
    if (tile0 >= cnt) return;
    int n0 = blockIdx.y * TNB;

    __shared__ float alds[2 * BUFSZ];
    __shared__ int   sidx[TM];

    int tid = threadIdx.x;
    if (tid < TM) {
        int r = tile0 + tid;
        sidx[tid] = lists[p * BATCH + (r < cnt ? r : cnt - 1)];
    }
    __syncthreads();

    // per-thread async-copy assignment: 4 threads per row, 8 x B128 per chunk
    int arow = tid >> 2;                  // 0..63
    int acol0 = (tid & 3) * 4;            // float4 index base within row
    const float* gsrc_row = x + (size_t)sidx[arow] * IN_DIM;

    int wave  = tid >> 5;
    int lane  = tid & 31;
    int col   = lane & 15;          // N within 16-wide subtile / C column
    int halfk = (lane >> 4) * 2;    // K sub-offset per fragment half
    int ncol  = n0 + wave * 16 + col;

    const float* wrow = W + ((size_t)p * OUT_DIM + ncol) * IN_DIM; // B column source
    float bv = bias[p * OUT_DIM + ncol];

    v8f acc[4];
#pragma unroll
    for (int rt = 0; rt < 4; ++rt)
#pragma unroll
        for (int v = 0; v < 8; ++v) acc[rt][v] = bv;   // fold bias into C

    // issue chunk 0 into buffer 0
#pragma unroll
    for (int j = 0; j < ISSUES; ++j) {
        int c = (acol0 + j * 16) * 4;     // float offset in row (16 float4 stride)
        __builtin_amdgcn_global_load_async_to_lds_b128(
            (AS1 v4i*)(gsrc_row + c),
            (AS3 v4i*)(&alds[arow * LDSS + c]), 0, 0);
    }

    for (int kc = 0; kc < NCHUNK; ++kc) {
        if (kc + 1 < NCHUNK) {
            // prefetch next chunk into the other buffer (read last at kc-1)
            const float* gs = gsrc_row + (kc + 1) * KC;
            float* ld = &alds[((kc + 1) & 1) * BUFSZ + arow * LDSS];
#pragma unroll
            for (int j = 0; j < ISSUES; ++j) {
                int c = (acol0 + j * 16) * 4;
                __builtin_amdgcn_global_load_async_to_lds_b128(
                    (AS1 v4i*)(gs + c), (AS3 v4i*)(ld + c), 0, 0);
            }
            __builtin_amdgcn_s_wait_asynccnt(ISSUES);  // chunk kc complete
        } else {
            __builtin_amdgcn_s_wait_asynccnt(0);
        }
        __syncthreads();

        const float* abase = &alds[(kc & 1) * BUFSZ];
        const float* wk = wrow + kc * KC;
#pragma unroll 4
        for (int ks = 0; ks < KC; ks += 4) {
            v2f bfrag = *(const v2f*)(wk + ks + halfk);
#pragma unroll
            for (int rt = 0; rt < 4; ++rt) {
                v2f afrag = *(const v2f*)(&abase[(rt * 16 + col) * LDSS + ks + halfk]);
                acc[rt] = __builtin_amdgcn_wmma_f32_16x16x4_f32(
                    false, afrag, false, bfrag, (short)0, acc[rt], false, false);
            }
        }
        __syncthreads();   // all waves done reading buf[kc&1] before it is refilled
    }

    // scatter-accumulate: out[sample, ncol] += 0.5 * acc   (2 experts/sample)
    int hi = (lane >> 4) * 8;
#pragma unroll
    for (int rt = 0; rt < 4; ++rt) {
#pragma unroll
        for (int v = 0; v < 8; ++v) {
            int r = rt * 16 + hi + v;
            if (tile0 + r < cnt) {
                int sb = sidx[r];
                __hip_atomic_fetch_add(out + (size_t)sb * OUT_DIM + ncol,
                                       0.5f * acc[rt][v],
                                       __ATOMIC_RELAXED, __HIP_MEMORY_SCOPE_AGENT);
            }
        }
    }
}

// ---------------------------------------------------------------------------
extern "C" void kernel_launch(void* const* d_in, const int* in_sizes, int n_in,
                              void* d_out, int out_size, void* d_ws, size_t ws_size,
                              hipStream_t stream) {
    const float* x  = (const float*)d_in[0];
    const float* W  = (const float*)d_in[1];
    const float* b  = (const float*)d_in[2];
    const float* Ws = (const float*)d_in[3];
    const float* bs = (const float*)d_in[4];
    float* out = (float*)d_out;

    int* counts = (int*)d_ws;                        // 8 ints
    int* lists  = (int*)((char*)d_ws + 1024);        // 8 x 8192 ints

    zero_kernel<<<2048, 256, 0, stream>>>(out, BATCH * OUT_DIM, counts);
    gate_kernel<<<BATCH / 8, 256, 0, stream>>>(x, Ws, bs, counts, lists);

    dim3 grid(BATCH / TM, OUT_DIM / TNB, NUM_POPS);  // worst case all samples on one expert
    moe_gemm_kernel<<<grid, 256, 0, stream>>>(x, W, b, counts, lists, out);
}